// SaliencyFeatureSuppression_75754633167196
// MI455X (gfx1250) — compile-verified
//
#include <hip/hip_runtime.h>
#include <hip/hip_bf16.h>

// Problem constants (from the reference): x is (B=64, H=64, W=64, C=192) f32.
#define B_DIM 64
#define H_DIM 64
#define W_DIM 64
#define C_DIM 192
#define HW (H_DIM * W_DIM)            // 4096 spatial positions per batch
#define TOPK 819                       // int(4096 * 0.2)
#define SUPPRESS_VALUE 0.1f

typedef __attribute__((ext_vector_type(2))) float v2f;
typedef __attribute__((ext_vector_type(8))) float v8f;

// ---------------------------------------------------------------------------
// Kernel 1: saliency = mean(|x|, axis=C) via V_WMMA_F32_16X16X4_F32.
// One wave reduces a tile of 16 spatial positions x 192 channels.
// A-matrix ISA layout (32-bit, 16x4): lanes 0-15 hold {K0,K1}, lanes 16-31
// hold {K2,K3} for rows M = lane%16.  B = ones -> D(m, n) = row-sum of A.
// C/D layout: lane 0 column holds M=0..7 in v0..v7; lane 16 holds M=8..15.
// ---------------------------------------------------------------------------
__global__ __launch_bounds__(128) void sfs_saliency_wmma(
    const float* __restrict__ x, float* __restrict__ sal) {
  const int lane = threadIdx.x & 31;
  const int wave = threadIdx.x >> 5;
  const int tile = blockIdx.x * 4 + wave;          // 0 .. B*HW/16 - 1
  const int row  = lane & 15;                      // M within tile
  const int hi   = lane >> 4;                      // 0 -> K0/K1, 1 -> K2/K3

  const float* base = x + (size_t)tile * (16 * C_DIM) + row * C_DIM + 2 * hi;

  v2f b; b.x = 1.0f; b.y = 1.0f;                   // B = ones (layout-invariant)
  v8f c = {};

#pragma unroll 8
  for (int k = 0; k < C_DIM; k += 4) {
    v2f a;
    a.x = __builtin_fabsf(base[k]);                // K = k + 2*hi
    a.y = __builtin_fabsf(base[k + 1]);            // K = k + 2*hi + 1
    // D = A x B + C  (v_wmma_f32_16x16x4_f32), accumulate row sums.
    c = __builtin_amdgcn_wmma_f32_16x16x4_f32(
        /*neg_a=*/false, a, /*neg_b=*/false, b,
        /*c_mod=*/(short)0, c, /*reuse_a=*/false, /*reuse_b=*/false);
  }

  const float scale = 1.0f / (float)C_DIM;
  if (lane == 0) {                                  // column N=0: rows 0..7
#pragma unroll
    for (int v = 0; v < 8; ++v) sal[tile * 16 + v] = c[v] * scale;
  } else if (lane == 16) {                          // column N=0: rows 8..15
#pragma unroll
    for (int v = 0; v < 8; ++v) sal[tile * 16 + 8 + v] = c[v] * scale;
  }
}

// ---------------------------------------------------------------------------
// Kernel 2: per-batch top-k threshold (radix select on float bits; sal >= 0
// so uint ordering == float ordering), then build the suppression mask.
// Scatter-with-clamp in the reference == gather over in-bounds 3x3 window.
// One 256-thread workgroup per batch; the 4096 saliency values live in LDS.
// ---------------------------------------------------------------------------
__global__ __launch_bounds__(256) void sfs_topk_mask(
    const float* __restrict__ sal, float* __restrict__ mask) {
  __shared__ unsigned s_sal[HW];      // 16 KB
  __shared__ unsigned s_hist[256];
  __shared__ unsigned s_sel[2];       // {chosen digit, remaining k}

  const int b   = blockIdx.x;
  const int tid = threadIdx.x;

  for (int i = tid; i < HW; i += 256)
    s_sal[i] = __float_as_uint(sal[b * HW + i]);
  __syncthreads();

  unsigned prefix = 0u, prefmask = 0u, kk = TOPK;
  for (int shift = 24; shift >= 0; shift -= 8) {
    s_hist[tid] = 0u;
    __syncthreads();
    for (int i = tid; i < HW; i += 256) {
      unsigned u = s_sal[i];
      if ((u & prefmask) == prefix)
        atomicAdd(&s_hist[(u >> shift) & 255u], 1u);
    }
    __syncthreads();
    if (tid == 0) {
      unsigned cum = 0u;
      int d = 255;
      for (; d > 0; --d) {
        unsigned h = s_hist[d];
        if (cum + h >= kk) break;
        cum += h;
      }
      s_sel[0] = (unsigned)d;
      s_sel[1] = kk - cum;
    }
    __syncthreads();
    prefix   |= s_sel[0] << shift;
    prefmask |= 255u << shift;
    kk        = s_sel[1];
    __syncthreads();
  }
  const unsigned thr = prefix;        // bit pattern of the k-th largest value

  for (int i = tid; i < HW; i += 256) {
    const int r = i >> 6, cc = i & 63;
    bool sup = false;
#pragma unroll
    for (int dr = -1; dr <= 1; ++dr) {
      const int nr = r + dr;
      if (nr < 0 || nr >= H_DIM) continue;
#pragma unroll
      for (int dc = -1; dc <= 1; ++dc) {
        const int nc = cc + dc;
        if (nc < 0 || nc >= W_DIM) continue;
        sup = sup || (s_sal[(nr << 6) + nc] >= thr);
      }
    }
    mask[b * HW + i] = sup ? SUPPRESS_VALUE : 1.0f;
  }
}

// ---------------------------------------------------------------------------
// Kernel 3: out = x * mask, broadcast over channels. float4 (B128) per lane;
// 192 % 4 == 0 so every float4 maps to exactly one mask value. Mask (1 MB)
// stays L2-resident. Streams 402 MB -> the dominant, bandwidth-bound pass.
// ---------------------------------------------------------------------------
__global__ __launch_bounds__(256) void sfs_apply(
    const float* __restrict__ x, const float* __restrict__ mask,
    float* __restrict__ out) {
  const size_t i4 = (size_t)blockIdx.x * 256 + threadIdx.x;
  const float4 v = ((const float4*)x)[i4];
  const float  m = mask[i4 / (C_DIM / 4)];
  float4 o;
  o.x = v.x * m; o.y = v.y * m; o.z = v.z * m; o.w = v.w * m;
  ((float4*)out)[i4] = o;
}

extern "C" void kernel_launch(void* const* d_in, const int* in_sizes, int n_in,
                              void* d_out, int out_size, void* d_ws, size_t ws_size,
                              hipStream_t stream) {
  (void)in_sizes; (void)n_in; (void)out_size; (void)ws_size;
  const float* x   = (const float*)d_in[0];
  float*       out = (float*)d_out;

  float* sal  = (float*)d_ws;               // B*HW floats = 1 MB
  float* mask = sal + (size_t)B_DIM * HW;   // B*HW floats = 1 MB

  // Pass 1: 16384 wave-tiles (16 positions each), 4 waves per block.
  sfs_saliency_wmma<<<(B_DIM * HW / 16) / 4, 128, 0, stream>>>(x, sal);
  // Pass 2: one workgroup per batch.
  sfs_topk_mask<<<B_DIM, 256, 0, stream>>>(sal, mask);
  // Pass 3: one float4 per thread, exact cover (50331648 / 4 / 256 blocks).
  const int total4 = B_DIM * HW * C_DIM / 4;
  sfs_apply<<<total4 / 256, 256, 0, stream>>>(x, mask, out);
}